// local_fuse_19739669692920
// MI455X (gfx1250) — compile-verified
//
#include <hip/hip_runtime.h>
#include <hip/hip_bf16.h>
#include <math.h>

// ---------------- problem constants (from reference setup) ----------------
#define BB   2
#define CC   128
#define CH   64      // C/2
#define CD   16      // C/8
#define NN   4096
#define KNN  20
#define BN_EPS 1e-5f

typedef float v2f __attribute__((ext_vector_type(2)));
typedef float v8f __attribute__((ext_vector_type(8)));
typedef unsigned v4u __attribute__((ext_vector_type(4)));
typedef int v4i __attribute__((ext_vector_type(4)));
typedef int v8i __attribute__((ext_vector_type(8)));

#if defined(__has_builtin)
#if __has_builtin(__builtin_amdgcn_tensor_load_to_lds)
#define HAVE_TDM 1
#endif
#endif

// =====================================================================
// TDM helper: 2D tile load Global -> LDS (f32 elements).
// D# packing per CDNA5 ISA ch.8.3/8.4:
//   group0: [1:0]=count=1, [63:32]=lds_addr, [120:64]=global_addr, [127:126]=type=2
//   group1: [17:16]=data_size=2 (4B), [79:48]=tensor_dim0, [111:80]=tensor_dim1,
//           [127:112]=tile_dim0, [143:128]=tile_dim1, [207:160]=tensor_dim0_stride
// LDS result: row-packed, row Y at lds_off + Y*tile_d0*4.
// =====================================================================
#ifdef HAVE_TDM
__device__ __forceinline__ void tdm_load_2d_f32(unsigned lds_off, const void* gptr,
                                                unsigned tensor_d0, unsigned tensor_d1,
                                                unsigned tile_d0, unsigned tile_d1,
                                                unsigned d0_stride) {
  unsigned long long ga = (unsigned long long)(uintptr_t)gptr;
  v4u g0;
  g0.x = 1u;                                        // count=1 (valid), user mode
  g0.y = lds_off;                                   // lds_addr (bytes)
  g0.z = (unsigned)(ga & 0xffffffffull);            // global_addr[31:0]
  g0.w = (unsigned)((ga >> 32) & 0x1ffffffull) | (2u << 30);  // addr[56:32] | type=2
  v8i g1;
  g1[0] = (int)(2u << 16);                          // data_size=4B; no mask/flags
  g1[1] = (int)((tensor_d0 & 0xffffu) << 16);       // tensor_dim0[15:0]
  g1[2] = (int)((tensor_d0 >> 16) | ((tensor_d1 & 0xffffu) << 16));
  g1[3] = (int)((tensor_d1 >> 16) | (tile_d0 << 16));  // tile_dim0
  g1[4] = (int)(tile_d1 & 0xffffu);                 // tile_dim1; tile_dim2=0
  g1[5] = (int)d0_stride;                           // tensor_dim0_stride[31:0]
  g1[6] = 0;                                        // stride hi / dim1_stride
  g1[7] = 0;
  v4i gz = {0, 0, 0, 0};
#if defined(__clang_major__) && (__clang_major__ >= 23)
  v8i gz8 = {0, 0, 0, 0, 0, 0, 0, 0};
  __builtin_amdgcn_tensor_load_to_lds(g0, g1, gz, gz, gz8, 0);
#else
  __builtin_amdgcn_tensor_load_to_lds(g0, g1, gz, gz, 0);
#endif
}
#endif

// =====================================================================
// K0: fold BN + bias into weights.
// =====================================================================
__global__ void prep_kernel(const float* __restrict__ w1, const float* __restrict__ b1,
                            const float* __restrict__ g1, const float* __restrict__ be1,
                            const float* __restrict__ m1, const float* __restrict__ v1,
                            const float* __restrict__ w2, const float* __restrict__ b2,
                            const float* __restrict__ g2, const float* __restrict__ be2,
                            const float* __restrict__ m2, const float* __restrict__ v2,
                            const float* __restrict__ bu, const float* __restrict__ gu,
                            const float* __restrict__ beu, const float* __restrict__ mu,
                            const float* __restrict__ vu,
                            float* __restrict__ P1, float* __restrict__ Q1, float* __restrict__ t1v,
                            float* __restrict__ Wp, float* __restrict__ t2v,
                            float* __restrict__ su, float* __restrict__ tu) {
  int c = threadIdx.x;
  if (c < CH) {
    float s1 = g1[c] / sqrtf(v1[c] + BN_EPS);
    t1v[c] = be1[c] + (b1[c] - m1[c]) * s1;
    for (int d = 0; d < 3; ++d) {
      P1[c * 3 + d] = s1 * w1[c * 6 + d];
      Q1[c * 3 + d] = s1 * (w1[c * 6 + 3 + d] - w1[c * 6 + d]);
    }
    float s2 = g2[c] / sqrtf(v2[c] + BN_EPS);
    t2v[c] = be2[c] + (b2[c] - m2[c]) * s2;
    for (int j = 0; j < CC; ++j) {
      float wa = w2[c * (2 * CC) + j];
      float wb = w2[c * (2 * CC) + CC + j];
      Wp[c * CC + j]        = s2 * wa;          // Gf rows
      Wp[(CH + c) * CC + j] = s2 * (wb - wa);   // Hf rows
    }
  }
  if (c < CC) {
    float s = gu[c] / sqrtf(vu[c] + BN_EPS);
    su[c] = s;
    tu[c] = beu[c] + (bu[c] - mu[c]) * s;
  }
}

// ===================== xx[b][n] = |xyz|^2 =====================
__global__ void xx_kernel(const float* __restrict__ xyz, float* __restrict__ xx) {
  int t = blockIdx.x * blockDim.x + threadIdx.x;
  if (t >= BB * NN) return;
  int b = t / NN, n = t % NN;
  float s = 0.f;
  for (int d = 0; d < 3; ++d) {
    float x = xyz[(b * 3 + d) * NN + n];
    s += x * x;
  }
  xx[t] = s;
}

// =====================================================================
// K1: KNN via f32 WMMA Gram tiles + streaming top-20.
// One wave per 16-row tile; pd = 2*<xi,xj> - |xi|^2 - |xj|^2.
// =====================================================================
__global__ void knn_kernel(const float* __restrict__ xyz, const float* __restrict__ xx,
                           int* __restrict__ topk) {
  const int gid = blockIdx.x;
  const int b   = gid / (NN / 16);
  const int i0  = (gid % (NN / 16)) * 16;
  const int lane = threadIdx.x;
  const int m = lane & 15, half = lane >> 4, kb = half * 2;

  v2f a;
  a.x = xyz[(b * 3 + kb) * NN + i0 + m];                          // kb in {0,2}
  a.y = (kb + 1 < 3) ? xyz[(b * 3 + kb + 1) * NN + i0 + m] : 0.f; // zero-pad K=3

  float xr[8];
#pragma unroll
  for (int r = 0; r < 8; ++r) xr[r] = xx[b * NN + i0 + r + 8 * half];

  float bestv[KNN];
  int   besti[KNN];
#pragma unroll
  for (int t = 0; t < KNN; ++t) { bestv[t] = -3.4e38f; besti[t] = 0x7fffffff; }

  __shared__ float tile[16 * 16];
  __shared__ float mv[32][KNN];
  __shared__ int   mi[32][KNN];

  for (int jt = 0; jt < NN / 16; ++jt) {
    const int j0 = jt * 16;
    v2f bf;
    bf.x = xyz[(b * 3 + kb) * NN + j0 + m];
    bf.y = (kb + 1 < 3) ? xyz[(b * 3 + kb + 1) * NN + j0 + m] : 0.f;

    v8f c = {};
    c = __builtin_amdgcn_wmma_f32_16x16x4_f32(false, a, false, bf, (short)0, c,
                                              false, false);
    float xxj = xx[b * NN + j0 + m];
    __syncthreads();
#pragma unroll
    for (int r = 0; r < 8; ++r)
      tile[(r + 8 * half) * 16 + m] = 2.f * c[r] - xr[r] - xxj;
    __syncthreads();

    // selection: lanes (m, m+16) share row m; each scans 8 columns
#pragma unroll
    for (int ccol = 0; ccol < 8; ++ccol) {
      int col = half * 8 + ccol;
      float val = tile[m * 16 + col];
      int j = j0 + col;
      if (val > bestv[KNN - 1]) {
        int p = KNN - 1;
        while (p > 0 && bestv[p - 1] < val) {
          bestv[p] = bestv[p - 1]; besti[p] = besti[p - 1]; --p;
        }
        bestv[p] = val; besti[p] = j;
      }
    }
  }

  // merge the two per-row sorted lists -> top 20, tie-break on index
#pragma unroll
  for (int t = 0; t < KNN; ++t) { mv[lane][t] = bestv[t]; mi[lane][t] = besti[t]; }
  __syncthreads();
  if (lane < 16) {
    int pa = 0, pb = 0;
    int outBase = (b * NN + i0 + lane) * KNN;
    for (int t = 0; t < KNN; ++t) {
      float va = mv[lane][pa], vb = mv[lane + 16][pb];
      int   ia = mi[lane][pa], ib = mi[lane + 16][pb];
      bool take_a = (va > vb) || (va == vb && ia < ib);
      if (take_a) { topk[outBase + t] = ia; ++pa; }
      else        { topk[outBase + t] = ib; ++pb; }
    }
  }
}

// ============ K2: xyz branch Gx/Hx (K=3, VALU) -> XH[b][n][0..127] ============
__global__ void gh_xyz_kernel(const float* __restrict__ xyz,
                              const float* __restrict__ P1, const float* __restrict__ Q1,
                              const float* __restrict__ t1v, float* __restrict__ XH) {
  int t = blockIdx.x * blockDim.x + threadIdx.x;
  if (t >= BB * NN) return;
  int b = t / NN, n = t % NN;
  float x0 = xyz[(b * 3 + 0) * NN + n];
  float x1 = xyz[(b * 3 + 1) * NN + n];
  float x2 = xyz[(b * 3 + 2) * NN + n];
  for (int c = 0; c < CH; ++c) {
    float g = P1[c * 3] * x0 + P1[c * 3 + 1] * x1 + P1[c * 3 + 2] * x2;
    float h = Q1[c * 3] * x0 + Q1[c * 3 + 1] * x1 + Q1[c * 3 + 2] * x2 + t1v[c];
    XH[(size_t)t * CC + c]      = g;
    XH[(size_t)t * CC + CH + c] = h;
  }
}

// =====================================================================
// K3: GH = Wp(128x128) @ F(128x4096) per batch, f32 WMMA.
// Block = 128 threads (4 waves), owns a 128(M) x 64(N) output panel.
// Feature panel 128(K) x 64(N) staged in LDS once per block via the
// Tensor Data Mover (TENSORcnt) -- fallback: cooperative vector loads.
// Wave w consumes its private 16-column slice; 8 M-tiles x 32 wmma each.
// Stored transposed as GH[b][n][c] so the 20-neighbor gather is contiguous.
// =====================================================================
__global__ void gemm_gh_kernel(const float* __restrict__ feat, const float* __restrict__ Wp,
                               const float* __restrict__ t2v, float* __restrict__ GH) {
  const int blk = blockIdx.x;
  const int b   = blk / (NN / 64);
  const int n0  = (blk % (NN / 64)) * 64;

  __shared__ float panel[CC * 64];   // 32 KB: [k][0..63]

#ifdef HAVE_TDM
  if (threadIdx.x < 32) {            // wave 0 issues the DMA
    unsigned lds_off = (unsigned)(uintptr_t)(void*)panel;  // LDS aperture low bits
    tdm_load_2d_f32(lds_off, feat + (size_t)b * CC * NN + n0,
                    /*tensor_d0=*/NN, /*tensor_d1=*/CC,
                    /*tile_d0=*/64, /*tile_d1=*/CC,
                    /*d0_stride=*/NN);
    __builtin_amdgcn_s_wait_tensorcnt(0);
  }
  __syncthreads();
#else
  for (int i = threadIdx.x; i < CC * 64; i += 128) {
    int row = i >> 6, col = i & 63;
    panel[i] = feat[((size_t)b * CC + row) * NN + n0 + col];
  }
  __syncthreads();
#endif

  const int wave = threadIdx.x >> 5;
  const int lane = threadIdx.x & 31;
  const int ln = lane & 15, half = lane >> 4, kb = half * 2;
  const int ncol = wave * 16;        // this wave's 16-column slice of the panel

  for (int mt = 0; mt < CC / 16; ++mt) {
    const float* Arow = Wp + (size_t)(mt * 16 + ln) * CC;
    v8f acc = {};
    for (int k = 0; k < CC; k += 4) {
      v2f a, bf;
      a.x = Arow[k + kb];
      a.y = Arow[k + kb + 1];
      bf.x = panel[(k + kb) * 64 + ncol + ln];
      bf.y = panel[(k + kb + 1) * 64 + ncol + ln];
      acc = __builtin_amdgcn_wmma_f32_16x16x4_f32(false, a, false, bf, (short)0, acc,
                                                  false, false);
    }
#pragma unroll
    for (int r = 0; r < 8; ++r) {
      int rloc = mt * 16 + r + 8 * half;   // 0..127
      int col  = n0 + ncol + ln;
      float v = acc[r] + (rloc >= CH ? t2v[rloc - CH] : 0.f);
      GH[((size_t)b * NN + col) * CC + rloc] = v;
    }
  }
}

// ============ K4: f_enc[b][n][c] = relu(max_k G[c, nbr] + H[c, n]) ============
__global__ void fenc_kernel(const float* __restrict__ XH, const float* __restrict__ GH,
                            const int* __restrict__ topk, float* __restrict__ fenc) {
  int bn = blockIdx.x;             // b*N + n
  int c  = threadIdx.x;            // 0..127
  __shared__ int nb[KNN];
  if (c < KNN) nb[c] = topk[bn * KNN + c];
  __syncthreads();
  const float* G = (c < CH) ? XH : GH;
  int cc = (c < CH) ? c : (c - CH);
  float mx = -3.4e38f;
  int bbase = (bn / NN) * NN;
#pragma unroll 4
  for (int k = 0; k < KNN; ++k)
    mx = fmaxf(mx, G[((size_t)(bbase + nb[k])) * CC + cc]);
  float h = G[(size_t)bn * CC + CH + cc];
  fenc[(size_t)bn * CC + c] = fmaxf(mx + h, 0.f);
}

// ============ K5: f1/f2 = relu(Wd @ f_enc) ============
__global__ void down_kernel(const float* __restrict__ fenc,
                            const float* __restrict__ wd1, const float* __restrict__ wd2,
                            float* __restrict__ f1, float* __restrict__ f2) {
  int t = blockIdx.x * blockDim.x + threadIdx.x;
  if (t >= BB * CD * NN) return;
  int b = t / (CD * NN), rem = t % (CD * NN);
  int o = rem / NN, n = rem % NN;
  const float* fr = fenc + ((size_t)b * NN + n) * CC;
  float s1 = 0.f, s2 = 0.f;
  for (int c = 0; c < CC; ++c) {
    float f = fr[c];
    s1 += wd1[o * CC + c] * f;
    s2 += wd2[o * CC + c] * f;
  }
  f1[t] = fmaxf(s1, 0.f);
  f2[t] = fmaxf(s2, 0.f);
}

// ============ K6: f_channel[o] = mean_n f1[0][o][n] ============
__global__ void fch_kernel(const float* __restrict__ f1, float* __restrict__ fch) {
  int o = blockIdx.x, tid = threadIdx.x;
  __shared__ float sbuf[256];
  float s = 0.f;
  for (int n = tid; n < NN; n += 256) s += f1[o * NN + n];
  sbuf[tid] = s;
  __syncthreads();
  for (int st = 128; st > 0; st >>= 1) {
    if (tid < st) sbuf[tid] += sbuf[tid + st];
    __syncthreads();
  }
  if (tid == 0) fch[o] = sbuf[0] / (float)NN;
}

// ============ K7: f_space[n] = mean_o f2[0][o][n] ============
__global__ void fsp_kernel(const float* __restrict__ f2, float* __restrict__ fsp) {
  int n = blockIdx.x * blockDim.x + threadIdx.x;
  if (n >= NN) return;
  float s = 0.f;
  for (int o = 0; o < CD; ++o) s += f2[o * NN + n];
  fsp[n] = s / (float)CD;
}

// ============ K8: final + up-projection + BN + relu + mish residual ============
__global__ void final_kernel(const float* __restrict__ fch, const float* __restrict__ fsp,
                             const float* __restrict__ f1, const float* __restrict__ f2,
                             const float* __restrict__ wup, const float* __restrict__ su,
                             const float* __restrict__ tu, const float* __restrict__ fenc,
                             float* __restrict__ out) {
  int bn = blockIdx.x;
  int b = bn / NN, n = bn % NN;
  int co = threadIdx.x;            // 0..127
  __shared__ float fin[CD];
  if (co < CD) {
    fin[co] = sqrtf(fch[co] * fsp[n] + 1e-12f)
            + f1[((size_t)b * CD + co) * NN + n]
            + f2[((size_t)b * CD + co) * NN + n];
  }
  __syncthreads();
  float u = 0.f;
#pragma unroll
  for (int c = 0; c < CD; ++c) u += wup[co * CD + c] * fin[c];
  u = fmaxf(u * su[co] + tu[co], 0.f);
  float x = fenc[(size_t)bn * CC + co] - u;
  float sp = (x > 20.f) ? x : log1pf(expf(x));
  out[((size_t)b * CC + co) * NN + n] = x * tanhf(sp);
}

// =====================================================================
extern "C" void kernel_launch(void* const* d_in, const int* in_sizes, int n_in,
                              void* d_out, int out_size, void* d_ws, size_t ws_size,
                              hipStream_t stream) {
  const float* xyz  = (const float*)d_in[0];
  const float* feat = (const float*)d_in[1];
  const float* w1   = (const float*)d_in[2];
  const float* b1   = (const float*)d_in[3];
  const float* g1   = (const float*)d_in[4];
  const float* be1  = (const float*)d_in[5];
  const float* m1   = (const float*)d_in[6];
  const float* v1   = (const float*)d_in[7];
  const float* w2   = (const float*)d_in[8];
  const float* b2   = (const float*)d_in[9];
  const float* g2   = (const float*)d_in[10];
  const float* be2  = (const float*)d_in[11];
  const float* m2   = (const float*)d_in[12];
  const float* v2   = (const float*)d_in[13];
  const float* wd1  = (const float*)d_in[14];
  const float* wd2  = (const float*)d_in[15];
  const float* wup  = (const float*)d_in[16];
  const float* bu   = (const float*)d_in[17];
  const float* gu   = (const float*)d_in[18];
  const float* beu  = (const float*)d_in[19];
  const float* mu   = (const float*)d_in[20];
  const float* vu   = (const float*)d_in[21];
  // d_in[22] = k (==20, hardcoded as KNN)
  float* out = (float*)d_out;

  // ---- carve workspace ----
  char* p = (char*)d_ws;
  auto carve = [&](size_t bytes) -> char* {
    char* r = p;
    p += (bytes + 255) & ~(size_t)255;
    return r;
  };
  float* xx   = (float*)carve((size_t)BB * NN * 4);
  int*   topk = (int*)  carve((size_t)BB * NN * KNN * 4);
  float* GH   = (float*)carve((size_t)BB * NN * CC * 4);
  float* XH   = (float*)carve((size_t)BB * NN * CC * 4);
  float* fenc = (float*)carve((size_t)BB * NN * CC * 4);
  float* f1   = (float*)carve((size_t)BB * CD * NN * 4);
  float* f2   = (float*)carve((size_t)BB * CD * NN * 4);
  float* fch  = (float*)carve(CD * 4);
  float* fsp  = (float*)carve(NN * 4);
  float* Wp   = (float*)carve((size_t)CC * CC * 4);
  float* t2v  = (float*)carve(CH * 4);
  float* P1   = (float*)carve(CH * 3 * 4);
  float* Q1   = (float*)carve(CH * 3 * 4);
  float* t1v  = (float*)carve(CH * 4);
  float* su   = (float*)carve(CC * 4);
  float* tu   = (float*)carve(CC * 4);
  if ((size_t)(p - (char*)d_ws) > ws_size) return;  // workspace too small

  prep_kernel<<<1, 128, 0, stream>>>(w1, b1, g1, be1, m1, v1,
                                     w2, b2, g2, be2, m2, v2,
                                     bu, gu, beu, mu, vu,
                                     P1, Q1, t1v, Wp, t2v, su, tu);

  xx_kernel<<<(BB * NN + 255) / 256, 256, 0, stream>>>(xyz, xx);

  knn_kernel<<<BB * (NN / 16), 32, 0, stream>>>(xyz, xx, topk);

  gh_xyz_kernel<<<(BB * NN + 127) / 128, 128, 0, stream>>>(xyz, P1, Q1, t1v, XH);

  gemm_gh_kernel<<<BB * (NN / 64), 128, 0, stream>>>(feat, Wp, t2v, GH);

  fenc_kernel<<<BB * NN, 128, 0, stream>>>(XH, GH, topk, fenc);

  down_kernel<<<(BB * CD * NN + 255) / 256, 256, 0, stream>>>(fenc, wd1, wd2, f1, f2);

  fch_kernel<<<CD, 256, 0, stream>>>(f1, fch);
  fsp_kernel<<<(NN + 255) / 256, 256, 0, stream>>>(f2, fsp);

  final_kernel<<<BB * NN, 128, 0, stream>>>(fch, fsp, f1, f2, wup, su, tu, fenc, out);
}